// YOLOLoss_29454885716158
// MI455X (gfx1250) — compile-verified
//
#include <hip/hip_runtime.h>

// YOLO loss for MI455X (gfx1250, wave32).
// Pipeline: Z(init maps) -> P(640 targets: IoU/argmax/scatter) ->
//           M(dense grid: noobj log term + sparse owner-cell terms,
//             reduced per-block with v_wmma_f32_16x16x4_f32) ->
//           F(deterministic final combine).
// Workspace use: ~1.38 MB (owner ints 1.04MB + noobj bytes 260KB +
//                640 records + 1014*16 block partials).

#define CELLS   259584      // 32*3*52*52
#define PER_B   8112        // 3*52*52
#define PER_A   2704        // 52*52
#define NCLS    80
#define NT      640         // 32*20
#define MBLOCKS 1014        // CELLS / 256 (exact)

typedef __attribute__((ext_vector_type(2))) float v2f;
typedef __attribute__((ext_vector_type(8))) float v8f;

__device__ __constant__ float c_aw[3] = {1.25f, 2.0f, 4.125f};   // ANCHORS/8
__device__ __constant__ float c_ah[3] = {1.625f, 3.75f, 2.875f};

struct Rec { float tx, ty, tw, th; int cls; int pad0, pad1, pad2; };

__device__ __forceinline__ float clip_log(float v) {
    return fmaxf(__logf(v), -100.0f);   // log(0) = -inf -> clipped to -100
}

// ---------------- init maps ----------------
__global__ void yolo_zero(int* __restrict__ owner, unsigned int* __restrict__ noobjW) {
    int i = blockIdx.x * blockDim.x + threadIdx.x;
    int stride = gridDim.x * blockDim.x;
    for (int k = i; k < CELLS; k += stride) owner[k] = -1;
    for (int k = i; k < CELLS / 4; k += stride) noobjW[k] = 0u;
}

// ---------------- per-target prep ----------------
__global__ void yolo_prep(const float* __restrict__ targets,
                          int* __restrict__ owner,
                          unsigned char* __restrict__ noobj,
                          Rec* __restrict__ recs) {
    int idx = blockIdx.x * blockDim.x + threadIdx.x;
    if (idx >= NT) return;
    const float* tp = targets + idx * 5;
    float c0 = tp[0], bx = tp[1], by = tp[2], bw = tp[3], bh = tp[4];
    bool valid = (c0 + bx + by + bw + bh) != 0.0f;
    float gx = bx * 52.0f, gy = by * 52.0f, gw = bw * 52.0f, gh = bh * 52.0f;
    int gi = (int)gx, gj = (int)gy;

    float areaG = (gw + 1.0f) * (gh + 1.0f);
    float iou[3];
#pragma unroll
    for (int aa = 0; aa < 3; ++aa) {
        float awv = c_aw[aa], ahv = c_ah[aa];
        float inter = fmaxf(fminf(gw, awv) + 1.0f, 0.0f) *
                      fmaxf(fminf(gh, ahv) + 1.0f, 0.0f);
        iou[aa] = inter / (areaG + (awv + 1.0f) * (ahv + 1.0f) - inter + 1e-16f);
    }
    int best = 0;                       // argmax, first-max wins (strict >)
    if (iou[1] > iou[0]) best = 1;
    if (iou[2] > iou[best]) best = 2;

    int b = idx / 20;
    bool inb = (gi >= 0) && (gi < 52) && (gj >= 0) && (gj < 52);
    if (valid && inb) {
        int cellB = ((b * 3 + best) * 52 + gj) * 52 + gi;
        atomicMax(&owner[cellB], idx);  // deterministic duplicate resolution
#pragma unroll
        for (int aa = 0; aa < 3; ++aa) {
            if (iou[aa] > 0.5f) {
                int cellA = ((b * 3 + aa) * 52 + gj) * 52 + gi;
                noobj[cellA] = 1;       // noobj_mask -> 0 here
            }
        }
    }
    Rec rec;
    rec.tx = gx - (float)gi;
    rec.ty = gy - (float)gj;
    rec.tw = __logf(gw / c_aw[best] + 1e-16f);
    rec.th = __logf(gh / c_ah[best] + 1e-16f);
    rec.cls = (int)c0;
    rec.pad0 = rec.pad1 = rec.pad2 = 0;
    recs[idx] = rec;
}

// ---------------- dense pass + WMMA block reduction ----------------
__global__ __launch_bounds__(256) void yolo_main(
    const float* __restrict__ pred,
    const int* __restrict__ owner,
    const unsigned char* __restrict__ noobj,
    const Rec* __restrict__ recs,
    float* __restrict__ blockParts) {
    int tid = threadIdx.x;
    int cell = blockIdx.x * 256 + tid;          // grid is exact: no tail
    int b = cell / PER_B;
    int r = cell - b * PER_B;
    int a = r / PER_A;
    const float* p = pred + (size_t)b * 689520 + (size_t)r * 85;

    float conf = p[4];
    float s0 = 0.f, s1 = 0.f, s2 = 0.f, s3 = 0.f, s4 = 0.f, s5 = 0.f, s6 = 0.f, s7 = 0.f;
    if (!noobj[cell]) s5 = -clip_log(1.0f - conf);          // noobj BCE term

    int ow = owner[cell];
    if (ow >= 0) {                                          // masked cell (<=640 total)
        Rec rec = recs[ow];
        float px = p[0] * 0.125f, py = p[1] * 0.125f;
        float x = px - floorf(px);
        float y = py - floorf(py);
        float w = __logf(p[2] * 0.125f / c_aw[a] + 1e-16f);
        float h = __logf(p[3] * 0.125f / c_ah[a] + 1e-16f);
        s0 = -(rec.tx * clip_log(x) + (1.0f - rec.tx) * clip_log(1.0f - x));
        s1 = -(rec.ty * clip_log(y) + (1.0f - rec.ty) * clip_log(1.0f - y));
        float dw = w - rec.tw, dh = h - rec.th;
        s2 = dw * dw;
        s3 = dh * dh;
        s4 = -clip_log(conf);                               // t=1 BCE term
        float cs = 0.0f;
#pragma unroll 4
        for (int c = 0; c < NCLS; ++c) {
            float pc = p[5 + c];
            cs -= (c == rec.cls) ? clip_log(pc) : clip_log(1.0f - pc);
        }
        s6 = cs;
        s7 = 1.0f;                                          // mask count
    }

    float s[8] = {s0, s1, s2, s3, s4, s5, s6, s7};
#pragma unroll
    for (int i = 0; i < 8; ++i)
#pragma unroll
        for (int off = 16; off > 0; off >>= 1)
            s[i] += __shfl_xor(s[i], off, 32);

    __shared__ float parts[64];                             // 8 waves x 8 accs
    int lane = tid & 31, wv = tid >> 5;
    if (lane == 0)
        for (int i = 0; i < 8; ++i) parts[wv * 8 + i] = s[i];
    __syncthreads();

    if (tid < 32) {   // wave-uniform branch: EXEC all ones for WMMA
        // A(16x4): A[i,k]=parts[k][i] (rows 0-7), A[8+i,k]=parts[4+k][i] (rows 8-15)
        // Layout: VGPR j, lanes 0-15 -> (M=lane, K=j); lanes 16-31 -> (M=lane-16, K=2+j)
        int row = tid & 15;
        int hi = tid >> 4;
        int accI = row & 7;
        int wbase = (row >> 3) * 4;
        v2f A;
        A.x = parts[(wbase + hi * 2 + 0) * 8 + accI];
        A.y = parts[(wbase + hi * 2 + 1) * 8 + accI];
        v2f Bm;
        Bm.x = 1.0f; Bm.y = 1.0f;                           // B = ones -> D row sums
        v8f C = {};
        v8f D = __builtin_amdgcn_wmma_f32_16x16x4_f32(
            false, A, false, Bm, (short)0, C, false, false);
        // D row sums replicated across columns:
        //   lane 0  holds rows 0-7  = acc sums over waves 0-3
        //   lane 16 holds rows 8-15 = acc sums over waves 4-7
        // Store both halves directly; final kernel adds them (deterministic).
        if ((tid & 15) == 0) {
            float* dst = blockParts + blockIdx.x * 16 + (tid >> 4) * 8;
#pragma unroll
            for (int i = 0; i < 8; ++i) dst[i] = D[i];
        }
    }
}

// ---------------- deterministic final combine ----------------
__global__ void yolo_final(const float* __restrict__ bp, float* __restrict__ out) {
    __shared__ float tot[8];
    int i = threadIdx.x;
    if (i < 8) {
        float s = 0.0f;
        for (int blk = 0; blk < MBLOCKS; ++blk)
            s += bp[blk * 16 + i] + bp[blk * 16 + 8 + i];
        tot[i] = s;
    }
    __syncthreads();
    if (i == 0) {
        const float N = 259584.0f;
        float loss = 2.5f * (tot[0] + tot[1]) / N      // LAMBDA_XY * (loss_x+loss_y)
                   + 2.5f * (tot[2] + tot[3]) / N      // LAMBDA_WH * (loss_w+loss_h)
                   + tot[4] / N + 0.5f * tot[5] / N    // loss_conf
                   + tot[6] / (tot[7] * 80.0f);        // loss_cls
        out[0] = loss;
    }
}

extern "C" void kernel_launch(void* const* d_in, const int* in_sizes, int n_in,
                              void* d_out, int out_size, void* d_ws, size_t ws_size,
                              hipStream_t stream) {
    (void)in_sizes; (void)n_in; (void)out_size; (void)ws_size;
    const float* pred = (const float*)d_in[0];
    const float* targets = (const float*)d_in[1];

    char* ws = (char*)d_ws;
    int* owner = (int*)(ws + 256);
    unsigned char* noobj = (unsigned char*)(ws + 256 + (size_t)CELLS * 4);
    Rec* recs = (Rec*)(ws + 256 + (size_t)CELLS * 4 + CELLS);
    float* blockParts = (float*)(ws + 256 + (size_t)CELLS * 4 + CELLS + NT * sizeof(Rec));

    yolo_zero<<<256, 256, 0, stream>>>(owner, (unsigned int*)noobj);
    yolo_prep<<<3, 256, 0, stream>>>(targets, owner, noobj, recs);
    yolo_main<<<MBLOCKS, 256, 0, stream>>>(pred, owner, noobj, recs, blockParts);
    yolo_final<<<1, 64, 0, stream>>>(blockParts, (float*)d_out);
}